// E23cCell_8151847928500
// MI455X (gfx1250) — compile-verified
//
#include <hip/hip_runtime.h>
#include <math.h>

// ---------------------------------------------------------------------------
// Problem constants (baked; chunk_size input is the constant 64)
// ---------------------------------------------------------------------------
#define BB    8
#define TT    2048
#define DD    1024
#define SS    64
#define CC    64          // chunk size
#define NCH   32          // TT/CC
#define NSL   16          // dim slices per batch
#define SLICE 64          // DD / NSL
#define WGS_PER_BATCH NSL

typedef __attribute__((ext_vector_type(2))) float v2f;
typedef __attribute__((ext_vector_type(8))) float v8f;

// ---------------------------------------------------------------------------
// Kernel 1: xw[m,n] = sum_k x[m,k] * Wx[n,k]   (M=16384, N=1024, K=1024)
// fp32 WMMA 16x16x4, 2M x 2N register tile per wave:
//   tiles (m0,n0) (m0,n1) (m0+16,n0) (m0+16,n1),  n1 = n0 + 512
// -> 4 float2 loads feed 4 wmma (1.0 loads/wmma), 4 independent accum chains.
//
// A 16x4 f32 frag: lane l<16 -> row l, VGPR0=K0,VGPR1=K1 ; lane l+16 -> row l,
// K2/K3.  B 4x16 mirrors it from Wx rows (B[k][n] = Wx[n][k]), so every lane
// loads one contiguous float2 from x and from Wx per k-step of 4.
// ---------------------------------------------------------------------------
__global__ void __launch_bounds__(256)
e23c_xw_gemm(const float* __restrict__ x, const float* __restrict__ Wx,
             float* __restrict__ out)
{
    const int lane = threadIdx.x & 31;
    const int wave = threadIdx.x >> 5;
    const int m0   = blockIdx.x * 32;                // M tiles m0, m0+16
    const int n0   = (blockIdx.y * 8 + wave) * 16;   // gridDim.y = 4
    const int n1   = n0 + 512;
    const int row  = lane & 15;
    const int koff = (lane >> 4) << 1;               // 0 or 2

    const float* ap0 = x  + (size_t)(m0 + row) * DD + koff;
    const float* ap1 = x  + (size_t)(m0 + 16 + row) * DD + koff;
    const float* bp0 = Wx + (size_t)(n0 + row) * DD + koff;
    const float* bp1 = Wx + (size_t)(n1 + row) * DD + koff;

    v8f c00 = {}, c01 = {}, c10 = {}, c11 = {};
#pragma unroll 4
    for (int k = 0; k < DD; k += 4) {
        v2f a0 = *(const v2f*)(ap0 + k);
        v2f a1 = *(const v2f*)(ap1 + k);
        v2f b0 = *(const v2f*)(bp0 + k);
        v2f b1 = *(const v2f*)(bp1 + k);
        c00 = __builtin_amdgcn_wmma_f32_16x16x4_f32(false, a0, false, b0,
                                                    (short)0, c00, false, false);
        c01 = __builtin_amdgcn_wmma_f32_16x16x4_f32(false, a0, false, b1,
                                                    (short)0, c01, false, false);
        c10 = __builtin_amdgcn_wmma_f32_16x16x4_f32(false, a1, false, b0,
                                                    (short)0, c10, false, false);
        c11 = __builtin_amdgcn_wmma_f32_16x16x4_f32(false, a1, false, b1,
                                                    (short)0, c11, false, false);
    }
    // C/D 16x16 layout: VGPR v, lanes 0-15 -> M=v ; lanes 16-31 -> M=v+8
    const int r   = (lane >> 4) * 8;
    const int col = lane & 15;
#pragma unroll
    for (int v = 0; v < 8; ++v) {
        out[(size_t)(m0 + v + r) * DD + n0 + col]      = c00[v];
        out[(size_t)(m0 + v + r) * DD + n1 + col]      = c01[v];
        out[(size_t)(m0 + 16 + v + r) * DD + n0 + col] = c10[v];
        out[(size_t)(m0 + 16 + v + r) * DD + n1 + col] = c11[v];
    }
}

// ---------------------------------------------------------------------------
// Per-batch spin barrier across the 16 workgroups of one batch.
// ---------------------------------------------------------------------------
struct __align__(64) Bar { unsigned count; unsigned phase; unsigned pad[14]; };

__device__ __forceinline__ void batch_barrier(Bar* bar, unsigned target)
{
    __threadfence();      // release: publish this block's global stores
    __syncthreads();
    if (threadIdx.x == 0) {
        unsigned prev = atomicAdd(&bar->count, 1u);
        if (prev == WGS_PER_BATCH - 1u) {
            __hip_atomic_store(&bar->count, 0u, __ATOMIC_RELAXED,
                               __HIP_MEMORY_SCOPE_AGENT);
            __threadfence();
            atomicAdd(&bar->phase, 1u);   // release
        } else {
            while (__hip_atomic_load(&bar->phase, __ATOMIC_ACQUIRE,
                                     __HIP_MEMORY_SCOPE_AGENT) < target) {
                __builtin_amdgcn_s_sleep(2);
            }
        }
    }
    __syncthreads();
    __threadfence();      // acquire: all threads invalidate before re-reading
}

// ---------------------------------------------------------------------------
// Kernel 2: persistent recurrence. Grid = 128 blocks = (batch b, slice k).
// Block (b,k) owns output dims e in [64k, 64k+64) and keeps:
//   - W_h rows [e-slice] x 1024      in LDS (256 KB, stride 1025 -> per-wave
//     bank = (j + d) % 64, conflict-free)
//   - tape[b][0..63][d-slice]        in LDS (stride 65)
//   - full h (1024 floats)           in LDS, re-pulled from L2 each step via
//     global_load_async_to_lds_b128 + s_wait_asynccnt (CDNA5 async path)
// Two per-batch barriers per step: (1) after partial attention scores,
// (2) after publishing the new h slice.
// ---------------------------------------------------------------------------
#define WHS_STRIDE 1025
#define TLS_STRIDE 65
#define HFULL_OFF_BYTES ((64 * WHS_STRIDE + SS * TLS_STRIDE) * 4)
#define SMEM_FLOATS (64 * WHS_STRIDE + SS * TLS_STRIDE + DD + SS + SS + 64 * 5 + 16)

// One b128 per lane: 256 threads x 16B = 4KB = full h vector, straight to LDS.
__device__ __forceinline__ void async_pull_h(const float* __restrict__ hglob_b)
{
    const unsigned lds_off = HFULL_OFF_BYTES + threadIdx.x * 16u;
    const float*   gsrc    = hglob_b + threadIdx.x * 4u;
    asm volatile("global_load_async_to_lds_b128 %0, %1, off scope:SCOPE_DEV"
                 :: "v"(lds_off), "v"(gsrc)
                 : "memory");
    asm volatile("s_wait_asynccnt 0" ::: "memory");
}

__global__ void __launch_bounds__(256, 1)
e23c_recur(float* __restrict__ out,      // [B,T,D]  in: xw, out: h_t (in place)
           float* __restrict__ tape,     // [B,S,D]  mutable tape (= h_tape out)
           float* __restrict__ hwork_out,// [B,D]
           float* __restrict__ hglob,    // ws: [B,D] current h
           float* __restrict__ spart,    // ws: [B,NSL,S] partial scores
           Bar*   __restrict__ bars,     // ws: [B]
           const float* __restrict__ Wh,
           const float* __restrict__ bh,
           const float* __restrict__ Wwr)
{
    const int blk = blockIdx.x;
    const int b   = blk / NSL;
    const int k   = blk % NSL;
    const int e0  = k * SLICE;
    const int tid = threadIdx.x;

    extern __shared__ float sm[];
    float* whs   = sm;                         // [64][1025]
    float* tls   = whs + 64 * WHS_STRIDE;      // [64][65]
    float* hfull = tls + SS * TLS_STRIDE;      // [1024]  (byte off = HFULL_OFF_BYTES)
    float* sc    = hfull + DD;                 // [64] scores -> exp
    float* attn  = sc + SS;                    // [64]
    float* gp    = attn + SS;                  // [64][5] gemv partials
    float* red   = gp + 64 * 5;                // [2] max, 1/sum

    // ---- one-time LDS residency ----
    for (int idx = tid; idx < 64 * DD; idx += 256) {
        int j = idx >> 10, d = idx & (DD - 1);
        whs[j * WHS_STRIDE + d] = Wh[(size_t)(e0 + j) * DD + d];
    }
    for (int idx = tid; idx < SS * SLICE; idx += 256) {
        int s = idx >> 6, d = idx & 63;
        tls[s * TLS_STRIDE + d] = tape[(size_t)(b * SS + s) * DD + e0 + d];
    }
    async_pull_h(hglob + b * DD);
    __syncthreads();

    Bar* bar = &bars[b];
    unsigned target = 0;
    const float inv_sqrt_d = 0.03125f;   // 1024^-0.5

    for (int c = 0; c < NCH; ++c) {
        for (int ts = 0; ts < CC; ++ts) {
            const int t = c * CC + ts;

            // ---- Phase A: partial scores ps[s] over our d-slice ----
            if (tid < SS) {
                const int s = tid;
                float acc = 0.f;
#pragma unroll 8
                for (int d = 0; d < SLICE; ++d)
                    acc += tls[s * TLS_STRIDE + d] * hfull[e0 + d];
                spart[(b * NSL + k) * SS + s] = acc;
            }
            batch_barrier(bar, ++target);

            // ---- Phase B: reduce scores, softmax ----
            if (tid < SS) {
                float acc = 0.f;
#pragma unroll
                for (int kk = 0; kk < NSL; ++kk)
                    acc += spart[(b * NSL + kk) * SS + tid];
                sc[tid] = acc * inv_sqrt_d;
            }
            __syncthreads();
            if (tid == 0) {
                float m = sc[0];
                for (int s = 1; s < SS; ++s) m = fmaxf(m, sc[s]);
                red[0] = m;
            }
            __syncthreads();
            if (tid < SS) sc[tid] = __expf(sc[tid] - red[0]);
            __syncthreads();
            if (tid == 0) {
                float sum = 0.f;
                for (int s = 0; s < SS; ++s) sum += sc[s];
                red[1] = 1.0f / sum;
            }
            __syncthreads();
            if (tid < SS) attn[tid] = sc[tid] * red[1];
            __syncthreads();

            // ---- GEMV slice: g[j] = sum_d h[d] * Wh[e0+j][d] ----
            {
                const int j = tid & 63, p = tid >> 6;  // p in [0,4)
                const float* wrow = &whs[j * WHS_STRIDE + p * 256];
                const float* hv   = &hfull[p * 256];
                float acc = 0.f;
#pragma unroll 8
                for (int d = 0; d < 256; ++d) acc += wrow[d] * hv[d];
                gp[j * 5 + p] = acc;
            }
            __syncthreads();

            // ---- combine: h_new = tanh(xw + gemv + read + b_h) ----
            if (tid < SS) {
                const int j = tid;
                float gem = gp[j * 5 + 0] + gp[j * 5 + 1] +
                            gp[j * 5 + 2] + gp[j * 5 + 3];
                float rd = 0.f;
#pragma unroll 8
                for (int s = 0; s < SS; ++s)
                    rd += attn[s] * tls[s * TLS_STRIDE + j];
                const size_t oidx = (size_t)(b * TT + t) * DD + e0 + j;
                float hn = tanhf(out[oidx] + gem + rd + bh[e0 + j]);
                out[oidx] = hn;                    // output h_t (in place)
                hglob[b * DD + e0 + j] = hn;       // publish slice
            }
            batch_barrier(bar, ++target);

            // ---- pull full updated h (async direct-to-LDS) ----
            async_pull_h(hglob + b * DD);
            __syncthreads();
        }

        // ---- chunk end: tape[b][c][e-slice] = W_write[e-slice,:] @ h ----
        {
            const int j = tid & 63, p = tid >> 6;
            const float* wrow = &Wwr[(size_t)(e0 + j) * DD + p * 256];
            const float* hv   = &hfull[p * 256];
            float acc = 0.f;
            __builtin_prefetch(wrow, 0, 1);        // global_prefetch_b8
#pragma unroll 8
            for (int d = 0; d < 256; ++d) acc += wrow[d] * hv[d];
            gp[j * 5 + p] = acc;
        }
        __syncthreads();
        if (tid < SS) {
            const int j = tid;
            float v = gp[j * 5 + 0] + gp[j * 5 + 1] +
                      gp[j * 5 + 2] + gp[j * 5 + 3];
            const int slot = c;                    // c % 64 == c (c < 32)
            tape[(size_t)(b * SS + slot) * DD + e0 + j] = v;
            tls[slot * TLS_STRIDE + j] = v;        // our slice == our dims
        }
        __syncthreads();
    }

    // ---- final h_work ----
    if (tid < SS) hwork_out[b * DD + e0 + tid] = hfull[e0 + tid];
}

// ---------------------------------------------------------------------------
// Host launcher. Inputs: 0 x_seq, 1 h_tape, 2 h_work, 3 W_h, 4 W_x, 5 b_h,
// 6 W_write, 7 chunk_size. d_out = output | h_tape | h_work concatenated.
// ---------------------------------------------------------------------------
extern "C" void kernel_launch(void* const* d_in, const int* in_sizes, int n_in,
                              void* d_out, int out_size, void* d_ws, size_t ws_size,
                              hipStream_t stream)
{
    (void)in_sizes; (void)n_in; (void)out_size; (void)ws_size;

    const float* x    = (const float*)d_in[0];
    const float* tin  = (const float*)d_in[1];
    const float* h0   = (const float*)d_in[2];
    const float* Wh   = (const float*)d_in[3];
    const float* Wx   = (const float*)d_in[4];
    const float* bh   = (const float*)d_in[5];
    const float* Wwr  = (const float*)d_in[6];

    float* OUT  = (float*)d_out;                               // B*T*D
    float* TAPE = OUT  + (size_t)BB * TT * DD;                 // B*S*D
    float* HWO  = TAPE + (size_t)BB * SS * DD;                 // B*D

    float* hglob = (float*)d_ws;                               // B*D
    float* spart = hglob + BB * DD;                            // B*NSL*S
    Bar*   bars  = (Bar*)(spart + BB * NSL * SS);              // B bars

    hipMemcpyAsync(TAPE, tin, sizeof(float) * BB * SS * DD,
                   hipMemcpyDeviceToDevice, stream);
    hipMemcpyAsync(hglob, h0, sizeof(float) * BB * DD,
                   hipMemcpyDeviceToDevice, stream);
    hipMemsetAsync(bars, 0, sizeof(Bar) * BB, stream);

    // GEMM: 512 blocks x 2 M-tiles; N covered by gridDim.y(4) x 8 waves x 2
    e23c_xw_gemm<<<dim3(512, 4), 256, 0, stream>>>(x, Wx, OUT);

    const size_t smem = SMEM_FLOATS * sizeof(float);           // ~285 KB < 320 KB
    e23c_recur<<<dim3(BB * NSL), 256, smem, stream>>>(
        OUT, TAPE, HWO, hglob, spart, bars, Wh, bh, Wwr);
}